// SOC_Solver_403726926277
// MI455X (gfx1250) — compile-verified
//
#include <hip/hip_runtime.h>
#include <math.h>

// SOCM const-M loss, MI455X (gfx1250, wave32).
// Two-phase design for HBM saturation:
//   Phase 1: contrib[t,b,:] for all 102400 (t,b) pairs, one wave each
//            (streams the 419MB nabla_b tensor with full-device MLP).
//   Phase 2: per-b reverse scan + fused diff = E @ sigma via
//            V_WMMA_F32_16X16X4_F32 (16 t-rows per tile), squared+weighted.
// Fallback: fully fused single-pass kernel if workspace is too small.

typedef __attribute__((ext_vector_type(2))) float v2f;
typedef __attribute__((ext_vector_type(8))) float v8f;

#define TSTEPS 200
#define NB 512
#define ND 32
#define NT (TSTEPS + 1)   // 201
#define ESTR 33           // padded LDS stride for E tile

// ---------------- workspace layout (bytes) ----------------
// [0,    2048)  : partials (512 floats)
// [2048, 6144)  : sigma_inv (1024 floats)
// [6144, ...)   : contrib C (200*512*32 floats = 13,107,200 B)
#define WS_PARTIALS_OFF 0
#define WS_SINV_OFF     2048
#define WS_C_OFF        6144
#define WS_FAST_BYTES   (WS_C_OFF + (size_t)TSTEPS * NB * ND * 4)

// =====================================================================
// Kernel 0: invert sigma (Gauss-Jordan, 1 block x 32 threads).
// sigma = I + 0.05*N is diagonally dominant -> no pivoting needed.
// =====================================================================
__global__ __launch_bounds__(32) void socm_invert_kernel(
    const float* __restrict__ sigma, float* __restrict__ sinv_out)
{
    __shared__ float aug[ND][2 * ND];
    const int tid = threadIdx.x;
    for (int c = 0; c < ND; ++c) {
        aug[tid][c]      = sigma[tid * ND + c];
        aug[tid][ND + c] = (tid == c) ? 1.0f : 0.0f;
    }
    __syncthreads();
    for (int p = 0; p < ND; ++p) {
        float inv = 1.0f / aug[p][p];
        aug[p][tid]      *= inv;
        aug[p][tid + ND] *= inv;
        __syncthreads();
        if (tid != p) {
            float f = aug[tid][p];
            for (int c = 0; c < 2 * ND; ++c) aug[tid][c] -= f * aug[p][c];
        }
        __syncthreads();
    }
    for (int r = 0; r < ND; ++r) sinv_out[r * ND + tid] = aug[r][ND + tid];
}

// =====================================================================
// Kernel 1 (heavy, memory-bound): contrib rows for all (t0,b) pairs.
//   u[k]  = sqrt(dt)*noise[k] + dt*control[k]
//   v     = sigma_inv^T @ u          (wave32 shuffle broadcasts)
//   C[t0,b,i] = dt*nabla_f[i] - nabla_b[t0,b,i,:] . v
// 8 waves / block, 1 pair / wave -> 102400 waves in flight.
// =====================================================================
__global__ __launch_bounds__(256) void socm_contrib_kernel(
    const float* __restrict__ ts,
    const float* __restrict__ noises,
    const float* __restrict__ controls,
    const float* __restrict__ nabla_f,
    const float* __restrict__ nabla_b,
    const float* __restrict__ sinv,     // (32,32): sinv[k][j]
    float* __restrict__ C)
{
    __shared__ float sInv[ND * ND];
    for (int idx = threadIdx.x; idx < ND * ND; idx += 256)
        sInv[idx] = sinv[idx];
    __syncthreads();

    const int lane = threadIdx.x & 31;
    const int wv   = threadIdx.x >> 5;
    const int pair = blockIdx.x * 8 + wv;     // 0 .. 200*512-1
    const int t0   = pair >> 9;               // /NB
    const int b    = pair & (NB - 1);

    const float dt  = ts[t0 + 1] - ts[t0];
    const float sdt = sqrtf(dt);
    const size_t base = ((size_t)t0 * NB + b) * ND;

    // u distributed across lanes
    const float u = sdt * noises[base + lane] + dt * controls[base + lane];

    // v[j] = sum_k sinv[k][j] * u[k]   (lane j holds v[j])
    float v = 0.0f;
    #pragma unroll
    for (int k = 0; k < ND; ++k)
        v = fmaf(__shfl(u, k, 32), sInv[k * ND + lane], v);

    // row i = lane of nabla_b[t0,b]: 8x b128 stream + shuffle-broadcast dot
    const float4* rowp = (const float4*)(nabla_b + (base + (size_t)lane) * ND);
    float s = 0.0f;
    #pragma unroll
    for (int q = 0; q < 8; ++q) {
        float4 m = rowp[q];
        s = fmaf(m.x, __shfl(v, 4 * q + 0, 32), s);
        s = fmaf(m.y, __shfl(v, 4 * q + 1, 32), s);
        s = fmaf(m.z, __shfl(v, 4 * q + 2, 32), s);
        s = fmaf(m.w, __shfl(v, 4 * q + 3, 32), s);
    }
    C[base + lane] = fmaf(dt, nabla_f[base + lane], -s);
}

// =====================================================================
// Kernel 2 (light): per-b reverse scan over t + WMMA loss tiles.
// Loads batched in 16-row chunks for MLP; only ~26MB total traffic.
// =====================================================================
__global__ __launch_bounds__(32) void socm_scan_fast_kernel(
    const float* __restrict__ nabla_V,
    const float* __restrict__ nabla_g,
    const float* __restrict__ sigma,
    const float* __restrict__ log_pd,
    const float* __restrict__ log_ps,
    const float* __restrict__ log_tw,
    const float* __restrict__ C,
    float* __restrict__ partials)
{
    __shared__ float sigLDS[ND * ND];
    __shared__ float Ebuf[16][ESTR];
    __shared__ float red[32];

    const int tid   = threadIdx.x;
    const int b     = blockIdx.x;
    const int row16 = tid & 15;
    const int hi    = tid >> 4;

    for (int idx = tid; idx < ND * ND; idx += 32) sigLDS[idx] = sigma[idx];

    float acc  = nabla_g[b * ND + tid];
    float lsum = 0.0f;

    auto do_tile = [&]() {
        __syncthreads();
        v8f d0 = {}; v8f d1 = {};
        #pragma unroll
        for (int q = 0; q < 8; ++q) {
            const int kk = 4 * q;
            v2f a, b0, b1;
            a.x  = Ebuf[row16][kk + 2 * hi + 0];
            a.y  = Ebuf[row16][kk + 2 * hi + 1];
            b0.x = sigLDS[(kk + 2 * hi + 0) * ND + row16];
            b0.y = sigLDS[(kk + 2 * hi + 1) * ND + row16];
            b1.x = sigLDS[(kk + 2 * hi + 0) * ND + 16 + row16];
            b1.y = sigLDS[(kk + 2 * hi + 1) * ND + 16 + row16];
            d0 = __builtin_amdgcn_wmma_f32_16x16x4_f32(false, a, false, b0,
                                                       (short)0, d0, false, false);
            d1 = __builtin_amdgcn_wmma_f32_16x16x4_f32(false, a, false, b1,
                                                       (short)0, d1, false, false);
        }
        #pragma unroll
        for (int r = 0; r < 8; ++r) lsum += d0[r] * d0[r] + d1[r] * d1[r];
        __syncthreads();
    };

    // 13 chunks of up to 16 t-rows, t descending from 200 to 0
    for (int chunk = 0; chunk < 13; ++chunk) {
        const int nrows = (chunk < 12) ? 16 : (NT - 12 * 16);   // 16 or 9
        float vV[16], cb[16];
        #pragma unroll
        for (int j = 0; j < 16; ++j) {          // batched loads (MLP)
            const int tj = TSTEPS - (chunk * 16 + j);
            if (j < nrows) {
                vV[j] = nabla_V[(size_t)tj * (NB * ND) + (size_t)b * ND + tid];
                cb[j] = (tj > 0)
                      ? C[((size_t)(tj - 1) * NB + b) * ND + tid] : 0.0f;
            }
        }
        #pragma unroll
        for (int j = 0; j < 16; ++j) {          // sequential scan (registers)
            if (j < nrows) {
                Ebuf[j][tid] = acc - vV[j];
                acc += cb[j];
            } else {
                Ebuf[j][tid] = 0.0f;
            }
        }
        do_tile();
    }

    const float w = expf(log_pd[b] + log_ps[b] + log_tw[b]);
    red[tid] = lsum * w;
    __syncthreads();
    if (tid == 0) {
        float ssum = 0.0f;
        for (int l = 0; l < 32; ++l) ssum += red[l];
        partials[b] = ssum;
    }
}

// =====================================================================
// Fallback: fully fused single-pass kernel (round-1 design); needs only
// the 2KB partials region of workspace.
// =====================================================================
__global__ __launch_bounds__(32) void socm_fused_kernel(
    const float* __restrict__ ts,
    const float* __restrict__ noises,
    const float* __restrict__ controls,
    const float* __restrict__ nabla_V,
    const float* __restrict__ nabla_f,
    const float* __restrict__ nabla_b,
    const float* __restrict__ nabla_g,
    const float* __restrict__ sigma,
    const float* __restrict__ log_pd,
    const float* __restrict__ log_ps,
    const float* __restrict__ log_tw,
    float* __restrict__ partials)
{
    __shared__ float sigLDS[ND * ND];
    __shared__ float aug[ND][2 * ND];
    __shared__ float Ebuf[16][ESTR];
    __shared__ float uvec[ND];
    __shared__ float vvec[ND];
    __shared__ float red[32];

    const int tid   = threadIdx.x;
    const int b     = blockIdx.x;
    const int row16 = tid & 15;
    const int hi    = tid >> 4;

    for (int idx = tid; idx < ND * ND; idx += 32) sigLDS[idx] = sigma[idx];
    __syncthreads();
    for (int c = 0; c < ND; ++c) {
        aug[tid][c]      = sigLDS[tid * ND + c];
        aug[tid][ND + c] = (tid == c) ? 1.0f : 0.0f;
    }
    __syncthreads();
    for (int p = 0; p < ND; ++p) {
        float inv = 1.0f / aug[p][p];
        aug[p][tid]      *= inv;
        aug[p][tid + ND] *= inv;
        __syncthreads();
        if (tid != p) {
            float f = aug[tid][p];
            for (int c = 0; c < 2 * ND; ++c) aug[tid][c] -= f * aug[p][c];
        }
        __syncthreads();
    }

    float acc  = nabla_g[b * ND + tid];
    float lsum = 0.0f;
    int   slot = 0;

    auto do_tile = [&]() {
        __syncthreads();
        v8f d0 = {}; v8f d1 = {};
        for (int q = 0; q < 8; ++q) {
            const int kk = 4 * q;
            v2f a, b0, b1;
            a.x  = Ebuf[row16][kk + 2 * hi + 0];
            a.y  = Ebuf[row16][kk + 2 * hi + 1];
            b0.x = sigLDS[(kk + 2 * hi + 0) * ND + row16];
            b0.y = sigLDS[(kk + 2 * hi + 1) * ND + row16];
            b1.x = sigLDS[(kk + 2 * hi + 0) * ND + 16 + row16];
            b1.y = sigLDS[(kk + 2 * hi + 1) * ND + 16 + row16];
            d0 = __builtin_amdgcn_wmma_f32_16x16x4_f32(false, a, false, b0,
                                                       (short)0, d0, false, false);
            d1 = __builtin_amdgcn_wmma_f32_16x16x4_f32(false, a, false, b1,
                                                       (short)0, d1, false, false);
        }
        for (int r = 0; r < 8; ++r) lsum += d0[r] * d0[r] + d1[r] * d1[r];
        __syncthreads();
    };

    for (int t = TSTEPS; t >= 0; --t) {
        float e = acc - nabla_V[(size_t)t * (NB * ND) + (size_t)b * ND + tid];
        Ebuf[slot][tid] = e;
        ++slot;
        if (slot == 16) { do_tile(); slot = 0; }

        if (t > 0) {
            const float dt  = ts[t] - ts[t - 1];
            const float sdt = sqrtf(dt);
            const size_t base = ((size_t)(t - 1) * NB + b) * ND;
            uvec[tid] = sdt * noises[base + tid] + dt * controls[base + tid];
            __syncthreads();
            float v = 0.0f;
            for (int k = 0; k < ND; ++k) v += uvec[k] * aug[k][ND + tid];
            vvec[tid] = v;
            __syncthreads();
            const float4* rowp =
                (const float4*)(nabla_b + (base + (size_t)tid) * ND);
            if (t > 1) {
                const char* np = (const char*)(nabla_b +
                    (((size_t)(t - 2) * NB + b) * ND + (size_t)tid) * ND);
                __builtin_prefetch(np, 0, 0);
                __builtin_prefetch(np + 64, 0, 0);
            }
            float s = 0.0f;
            for (int q = 0; q < 8; ++q) {
                float4 m = rowp[q];
                s += m.x * vvec[4 * q + 0] + m.y * vvec[4 * q + 1]
                   + m.z * vvec[4 * q + 2] + m.w * vvec[4 * q + 3];
            }
            acc += dt * nabla_f[base + tid] - s;
            __syncthreads();
        }
    }
    if (slot > 0) {
        for (int r = slot; r < 16; ++r) Ebuf[r][tid] = 0.0f;
        do_tile();
    }

    const float w = expf(log_pd[b] + log_ps[b] + log_tw[b]);
    red[tid] = lsum * w;
    __syncthreads();
    if (tid == 0) {
        float ssum = 0.0f;
        for (int l = 0; l < 32; ++l) ssum += red[l];
        partials[b] = ssum;
    }
}

__global__ void socm_finalize(const float* __restrict__ partials,
                              float* __restrict__ out)
{
    if (threadIdx.x == 0 && blockIdx.x == 0) {
        float s = 0.0f;
        for (int i = 0; i < NB; ++i) s += partials[i];   // fixed order
        out[0] = s / (float)(NT * NB);
    }
}

extern "C" void kernel_launch(void* const* d_in, const int* in_sizes, int n_in,
                              void* d_out, int out_size, void* d_ws, size_t ws_size,
                              hipStream_t stream) {
    const float* ts       = (const float*)d_in[0];
    // d_in[1] = states (unused by the loss)
    const float* noises   = (const float*)d_in[2];
    const float* controls = (const float*)d_in[3];
    const float* nabla_V  = (const float*)d_in[4];
    const float* nabla_f  = (const float*)d_in[5];
    const float* nabla_b  = (const float*)d_in[6];
    const float* nabla_g  = (const float*)d_in[7];
    const float* sigma    = (const float*)d_in[8];
    const float* log_pd   = (const float*)d_in[9];
    const float* log_ps   = (const float*)d_in[10];
    const float* log_tw   = (const float*)d_in[11];

    char*  ws       = (char*)d_ws;
    float* partials = (float*)(ws + WS_PARTIALS_OFF);

    if (ws_size >= WS_FAST_BYTES) {
        float* sinv = (float*)(ws + WS_SINV_OFF);
        float* C    = (float*)(ws + WS_C_OFF);
        socm_invert_kernel<<<1, 32, 0, stream>>>(sigma, sinv);
        socm_contrib_kernel<<<(TSTEPS * NB) / 8, 256, 0, stream>>>(
            ts, noises, controls, nabla_f, nabla_b, sinv, C);
        socm_scan_fast_kernel<<<NB, 32, 0, stream>>>(
            nabla_V, nabla_g, sigma, log_pd, log_ps, log_tw, C, partials);
    } else {
        socm_fused_kernel<<<NB, 32, 0, stream>>>(
            ts, noises, controls, nabla_V, nabla_f, nabla_b, nabla_g, sigma,
            log_pd, log_ps, log_tw, partials);
    }
    socm_finalize<<<1, 1, 0, stream>>>(partials, (float*)d_out);
}